// GAT_26036091748990
// MI455X (gfx1250) — compile-verified
//
#include <hip/hip_runtime.h>
#include <math.h>

// GAT, B=2, N=512, IN=512, MEM=300 (padded 304), HID=64, 2 layers.
// All GEMMs via V_WMMA_F32_16X16X4_F32 (fp32-exact matrix pipe).
// Edge-score pass fused (never materializes the (B,N,N,H) z tensor).
// Global-softmax 1/sum folded into the aggregation GEMM epilogue.

#define BB    2
#define NN    512
#define IND   512
#define MEMD  300
#define PADD  304
#define HIDD  64
#define WG    256
#define RBLK  128   // partial-reduction blocks per batch

typedef float v2f __attribute__((ext_vector_type(2)));
typedef float v8f __attribute__((ext_vector_type(8)));

// ---------------------------------------------------------------------------
// fp32 WMMA GEMM: C[M x Nw] = scale[b] * (A[M x K] @ W[K x Nw]) + bias
//   - one wave per 16x16 tile, K stepped by 4 (v_wmma_f32_16x16x4_f32)
//   - A layout (ISA 7.12.2): lane l holds A[m0 + l%16][k0 + 2*(l/16) + v]
//   - B layout (transpose-symmetric): lane l holds W[k0 + 2*(l/16) + v][n0 + l%16]
//   - C/D layout: vgpr v, lane l -> C[m0 + v + 8*(l/16)][n0 + l%16]
//   - OOB columns are CLAMPED (not predicated): garbage only lands in columns
//     that the store guard discards, so the inner loop stays branch-free and
//     EXEC stays all-1s (WMMA requirement).
//   - LDA/LDW/K are compile-time so address math folds into load offsets.
// ---------------------------------------------------------------------------
template<int LDA, int LDW, int K>
__global__ void wmma_gemm_f32(const float* __restrict__ A,
                              const float* __restrict__ W, long wBatchStride,
                              const float* __restrict__ bias,
                              const float* __restrict__ scale,
                              float* __restrict__ C, int ldc,
                              int M, int Nw, int Ntiles, int rowsPerBatch)
{
    int wave = (int)((blockIdx.x * blockDim.x + threadIdx.x) >> 5);
    int lane = (int)(threadIdx.x & 31u);
    int mt = wave / Ntiles;
    int nt = wave - mt * Ntiles;
    if (mt * 16 >= M) return;                 // wave-uniform exit

    int m0   = mt * 16;
    int n0   = nt * 16;
    int r16  = lane & 15;                     // A-row / B,D-col sub-index
    int kh   = (lane >> 4) << 1;              // 0 or 2
    int half = (lane >> 4) << 3;              // 0 or 8 (D-row offset)
    int b    = m0 / rowsPerBatch;
    int cg   = n0 + r16;
    int cgc  = cg < Nw ? cg : (Nw - 1);       // clamp instead of predicate

    const float* aPtr = A + (long)(m0 + r16) * LDA + kh;
    const float* bPtr = W + (long)b * wBatchStride + (long)kh * LDW + cgc;

    v8f acc = {0.f, 0.f, 0.f, 0.f, 0.f, 0.f, 0.f, 0.f};

#pragma unroll 4
    for (int k0 = 0; k0 < K; k0 += 4) {
        v2f a = *(const v2f*)aPtr;            // global_load_b64 (8B aligned)
        v2f bf;
        bf.x = bPtr[0];
        bf.y = bPtr[LDW];
        acc = __builtin_amdgcn_wmma_f32_16x16x4_f32(
                  false, a, false, bf, (short)0, acc, false, false);
        aPtr += 4;
        bPtr += 4 * (long)LDW;
    }

    if (cg < Nw) {
        float s  = scale ? scale[b] : 1.0f;
        float bv = bias  ? bias[cg] : 0.0f;
        long  rbase = m0 + half;
#pragma unroll
        for (int v = 0; v < 8; ++v)
            C[(rbase + v) * (long)ldc + cg] = acc[v] * s + bv;
    }
}

template<int LDA, int LDW, int K>
static void launch_gemm(const float* A, const float* W, long wbs,
                        const float* bias, const float* scale,
                        float* C, int ldc, int M, int Nw, hipStream_t stream)
{
    int Ntiles = (Nw + 15) / 16;
    int waves  = (M / 16) * Ntiles;
    int blocks = (waves * 32 + WG - 1) / WG;
    hipLaunchKernelGGL((wmma_gemm_f32<LDA, LDW, K>), dim3(blocks), dim3(WG), 0,
                       stream, A, W, wbs, bias, scale, C, ldc, M, Nw, Ntiles, NN);
}

// ---------------------------------------------------------------------------
// Fused attention scores (no (B,N,N,H) tensor):
//   e[b,i,j] = leakyrelu( sum_h relu(si[b,i,h] + sj[b,j,h]) * a2w[h] + a2b )
//   logits   = adj ? e : -1e30          (a1_b is pre-folded into si)
// ---------------------------------------------------------------------------
__global__ void gat_edge_scores(const float* __restrict__ si,
                                const float* __restrict__ sj,
                                const float* __restrict__ a2w,
                                const float* __restrict__ a2b,
                                const float* __restrict__ adj,
                                float* __restrict__ e)
{
    int bi = (int)blockIdx.x;                 // 0 .. B*N-1
    int b  = bi >> 9;                         // / 512
    __shared__ float s_si[HIDD];
    __shared__ float s_w[HIDD];
    if (threadIdx.x < HIDD) {
        s_si[threadIdx.x] = si[(size_t)bi * HIDD + threadIdx.x];
        s_w[threadIdx.x]  = a2w[threadIdx.x];
    }
    __syncthreads();

    float a2bv = a2b[0];
    const float*  adjRow = adj + (size_t)bi * NN;
    float*        eRow   = e   + (size_t)bi * NN;
    const float*  sjB    = sj  + (size_t)b * NN * HIDD;

    for (int j = (int)threadIdx.x; j < NN; j += WG) {
        const float4* sjr = (const float4*)(sjB + (size_t)j * HIDD);
        float acc = 0.f;
#pragma unroll
        for (int h4 = 0; h4 < HIDD / 4; ++h4) {
            float4 v = sjr[h4];
            float z0 = s_si[4*h4+0] + v.x;  z0 = z0 > 0.f ? z0 : 0.f;
            float z1 = s_si[4*h4+1] + v.y;  z1 = z1 > 0.f ? z1 : 0.f;
            float z2 = s_si[4*h4+2] + v.z;  z2 = z2 > 0.f ? z2 : 0.f;
            float z3 = s_si[4*h4+3] + v.w;  z3 = z3 > 0.f ? z3 : 0.f;
            acc = fmaf(z0, s_w[4*h4+0], acc);
            acc = fmaf(z1, s_w[4*h4+1], acc);
            acc = fmaf(z2, s_w[4*h4+2], acc);
            acc = fmaf(z3, s_w[4*h4+3], acc);
        }
        float ev = acc + a2bv;
        ev = ev > 0.f ? ev : ev * 0.01f;                 // leaky relu
        eRow[j] = (adjRow[j] != 0.f) ? ev : -1e30f;      // adj mask (adj in {0,1})
    }
}

// ---------------------------------------------------------------------------
// Global softmax reductions over the flattened N*N logits per batch
// ---------------------------------------------------------------------------
__global__ void reduce_max_partial(const float* __restrict__ e,
                                   float* __restrict__ partials,
                                   int perBatch, int nblocks)
{
    int b = (int)blockIdx.y;
    const float* base = e + (size_t)b * perBatch;
    float m = -3.4e38f;
    for (int i = (int)(blockIdx.x * blockDim.x + threadIdx.x); i < perBatch;
         i += WG * nblocks)
        m = fmaxf(m, base[i]);
    __shared__ float red[WG];
    red[threadIdx.x] = m;
    __syncthreads();
    for (int s = WG / 2; s > 0; s >>= 1) {
        if ((int)threadIdx.x < s)
            red[threadIdx.x] = fmaxf(red[threadIdx.x], red[threadIdx.x + s]);
        __syncthreads();
    }
    if (threadIdx.x == 0) partials[(size_t)b * nblocks + blockIdx.x] = red[0];
}

__global__ void exp_sum_partial(float* __restrict__ e,
                                const float* __restrict__ maxv,
                                float* __restrict__ partials,
                                int perBatch, int nblocks)
{
    int b = (int)blockIdx.y;
    float m = maxv[b];
    float* base = e + (size_t)b * perBatch;
    float s = 0.f;
    for (int i = (int)(blockIdx.x * blockDim.x + threadIdx.x); i < perBatch;
         i += WG * nblocks) {
        float v = __expf(base[i] - m);   // masked logits (-1e30) underflow to 0
        base[i] = v;
        s += v;
    }
    __shared__ float red[WG];
    red[threadIdx.x] = s;
    __syncthreads();
    for (int st = WG / 2; st > 0; st >>= 1) {
        if ((int)threadIdx.x < st)
            red[threadIdx.x] += red[threadIdx.x + st];
        __syncthreads();
    }
    if (threadIdx.x == 0) partials[(size_t)b * nblocks + blockIdx.x] = red[0];
}

// doInvSum=0 -> out[b] = max(partials);  doInvSum=1 -> out[b] = 1/sum(partials)
__global__ void reduce_final(const float* __restrict__ partials,
                             float* __restrict__ out, int n, int doInvSum)
{
    int b = (int)blockIdx.x;
    const float* p = partials + (size_t)b * n;
    float v = doInvSum ? 0.f : -3.4e38f;
    for (int i = (int)threadIdx.x; i < n; i += WG)
        v = doInvSum ? (v + p[i]) : fmaxf(v, p[i]);
    __shared__ float red[WG];
    red[threadIdx.x] = v;
    __syncthreads();
    for (int s = WG / 2; s > 0; s >>= 1) {
        if ((int)threadIdx.x < s)
            red[threadIdx.x] = doInvSum ? (red[threadIdx.x] + red[threadIdx.x + s])
                                        : fmaxf(red[threadIdx.x], red[threadIdx.x + s]);
        __syncthreads();
    }
    if (threadIdx.x == 0) out[b] = doInvSum ? (1.0f / red[0]) : red[0];
}

// ---------------------------------------------------------------------------
extern "C" void kernel_launch(void* const* d_in, const int* in_sizes, int n_in,
                              void* d_out, int out_size, void* d_ws, size_t ws_size,
                              hipStream_t stream)
{
    (void)in_sizes; (void)n_in; (void)out_size; (void)ws_size;
    const float* feature = (const float*)d_in[0];  // (B,N,512)
    const float* adj     = (const float*)d_in[1];  // (B,N,N)
    const float* w0      = (const float*)d_in[2];  // (512,300)
    const float* b0      = (const float*)d_in[3];  // (300,)
    const float* w1      = (const float*)d_in[4];  // (300,300)
    const float* b1      = (const float*)d_in[5];  // (300,)
    const float* a1w     = (const float*)d_in[6];  // (600,64)
    const float* a1b     = (const float*)d_in[7];  // (64,)
    const float* a2w     = (const float*)d_in[8];  // (64,1)
    const float* a2b     = (const float*)d_in[9];  // (1,)
    float* out = (float*)d_out;                    // (B,N,300)

    // workspace layout (floats); total ~1.28M floats (~5.1 MB)
    float* ws   = (float*)d_ws;
    float* H    = ws;  ws += (size_t)BB * NN * PADD;   // h, padded cols
    float* SI   = ws;  ws += (size_t)BB * NN * HIDD;
    float* SJ   = ws;  ws += (size_t)BB * NN * HIDD;
    float* E    = ws;  ws += (size_t)BB * NN * NN;     // logits -> exp in place
    float* O1   = ws;  ws += (size_t)BB * NN * PADD;   // layer-1 output, padded
    float* PART = ws;  ws += (size_t)BB * RBLK;
    float* MAXV = ws;  ws += BB;
    float* INV  = ws;  ws += BB;

    auto softmax_and_aggregate = [&](float* OUT, int ldc) {
        hipLaunchKernelGGL(gat_edge_scores, dim3(BB * NN), dim3(WG), 0, stream,
                           SI, SJ, a2w, a2b, adj, E);
        hipLaunchKernelGGL(reduce_max_partial, dim3(RBLK, BB), dim3(WG), 0, stream,
                           E, PART, NN * NN, RBLK);
        hipLaunchKernelGGL(reduce_final, dim3(BB), dim3(WG), 0, stream,
                           PART, MAXV, RBLK, 0);
        hipLaunchKernelGGL(exp_sum_partial, dim3(RBLK, BB), dim3(WG), 0, stream,
                           E, MAXV, PART, NN * NN, RBLK);
        hipLaunchKernelGGL(reduce_final, dim3(BB), dim3(WG), 0, stream,
                           PART, INV, RBLK, 1);
        // out = (1/sum) * (E @ h): batched B matrix, per-batch scale
        launch_gemm<NN, PADD, NN>(E, H, (long)NN * PADD, nullptr, INV,
                                  OUT, ldc, BB * NN, MEMD, stream);
    };

    // ---- layer 1 ----
    launch_gemm<IND, MEMD, IND>(feature, w0, 0, b0, nullptr,
                                H, PADD, BB * NN, MEMD, stream);
    launch_gemm<PADD, HIDD, MEMD>(H, a1w, 0, a1b, nullptr,
                                  SI, HIDD, BB * NN, HIDD, stream);
    launch_gemm<PADD, HIDD, MEMD>(H, a1w + MEMD * HIDD, 0, nullptr, nullptr,
                                  SJ, HIDD, BB * NN, HIDD, stream);
    softmax_and_aggregate(O1, PADD);

    // ---- layer 2 ----
    launch_gemm<PADD, MEMD, MEMD>(O1, w1, 0, b1, nullptr,
                                  H, PADD, BB * NN, MEMD, stream);
    launch_gemm<PADD, HIDD, MEMD>(H, a1w, 0, a1b, nullptr,
                                  SI, HIDD, BB * NN, HIDD, stream);
    launch_gemm<PADD, HIDD, MEMD>(H, a1w + MEMD * HIDD, 0, nullptr, nullptr,
                                  SJ, HIDD, BB * NN, HIDD, stream);
    softmax_and_aggregate(out, MEMD);
}